// LpNormPool2d_63402307224335
// MI455X (gfx1250) — compile-verified
//
#include <hip/hip_runtime.h>

// LpNormPool2d, kernel=stride=2x2, fp32.
// B=16, C=256, H=W=112, Ho=Wo=56. HBM-bound: 257 MB traffic -> ~11 us floor.
// CDNA5 path: GLOBAL_LOAD_ASYNC_TO_LDS_B128 double-buffered per wave
// (ASYNCcnt software pipeline), ds_load_b128 readback, NT b64 stores.

typedef float v4f __attribute__((ext_vector_type(4)));
typedef float v2f __attribute__((ext_vector_type(2)));

#define BDIM  256
#define NWAVE 8
#define C_    256
#define H_    112
#define W_    112
#define HO    56
#define WO    56

// |d|^p via hardware log2/exp2 (v_log_f32 / v_exp_f32).
// d == 0: log2 -> -inf -> exp2 -> 0 (matches 0^p, p>0).
__device__ __forceinline__ float pow_abs(float d, float p) {
    return __builtin_amdgcn_exp2f(p * __builtin_amdgcn_logf(__builtin_fabsf(d)));
}

__global__ __launch_bounds__(BDIM) void lp_pool2d_async_kernel(
    const float* __restrict__ x,   // (B,C,H,W)
    const float* __restrict__ pp,  // (C,)
    const float* __restrict__ cc,  // (C,2,2)
    float* __restrict__ out)       // (B,C,Ho,Wo)
{
    // Per-wave private double buffer: [wave][stage][row-in-pair][cols(pad 128)]
    __shared__ float lds[NWAVE][2][2][128];   // 16 KB total

    const int bc = blockIdx.x;                // b*C + ch
    const int ch = bc & (C_ - 1);
    const float* __restrict__ xb = x + (size_t)bc * (H_ * W_);
    float* __restrict__ ob = out + (size_t)bc * (HO * WO);

    // Block-uniform per-channel parameters -> scalar loads.
    const float p   = pp[ch];
    const float ip  = __builtin_amdgcn_rcpf(p);   // exponent 1/p, rcp is plenty
    const float c00 = cc[ch * 4 + 0];
    const float c01 = cc[ch * 4 + 1];
    const float c10 = cc[ch * 4 + 2];
    const float c11 = cc[ch * 4 + 3];

    const int w    = threadIdx.x >> 5;   // wave id 0..7
    const int lane = threadIdx.x & 31;
    const bool act = lane < 28;          // 28 lanes x 16B = one 448B input row

    // Stage input rows (2r, 2r+1) for output row r into buffer b:
    // two async b128 copies per active lane, non-temporal (zero reuse).
    auto issue = [&](int r, int b) {
        if (act) {
            const float* g0 = xb + (2 * r) * W_ + lane * 4;
            unsigned l0 = (unsigned)(uintptr_t)&lds[w][b][0][lane * 4];
            unsigned l1 = (unsigned)(uintptr_t)&lds[w][b][1][lane * 4];
            asm volatile("global_load_async_to_lds_b128 %0, %1, off th:TH_LOAD_NT"
                         :: "v"(l0), "v"((unsigned long long)(uintptr_t)g0)
                         : "memory");
            asm volatile("global_load_async_to_lds_b128 %0, %1, off th:TH_LOAD_NT"
                         :: "v"(l1), "v"((unsigned long long)(uintptr_t)(g0 + W_))
                         : "memory");
        }
    };

    issue(w, 0);                          // prime the pipeline

    for (int i = 0; i < 7; ++i) {         // wave w handles rows w, w+8, ..., w+48
        const int r = w + 8 * i;
        const int b = i & 1;

        if (i < 6) {
            issue(r + 8, b ^ 1);          // copies run 1 stage ahead of compute
            // Async loads complete in order per wave: <=2 outstanding means
            // the 2 copies feeding buffer b are done.
            asm volatile("s_wait_asynccnt 0x2" ::: "memory");
        } else {
            asm volatile("s_wait_asynccnt 0x0" ::: "memory");
        }

        if (act) {
            v4f a  = *(const v4f*)&lds[w][b][0][lane * 4];  // ds_load_b128
            v4f bb = *(const v4f*)&lds[w][b][1][lane * 4];

            const float s0 = pow_abs(a.x - c00, p) + pow_abs(a.y - c01, p)
                           + pow_abs(bb.x - c10, p) + pow_abs(bb.y - c11, p);
            const float s1 = pow_abs(a.z - c00, p) + pow_abs(a.w - c01, p)
                           + pow_abs(bb.z - c10, p) + pow_abs(bb.w - c11, p);

            v2f o;
            o.x = __builtin_amdgcn_exp2f(ip * __builtin_amdgcn_logf(s0 * 0.25f));
            o.y = __builtin_amdgcn_exp2f(ip * __builtin_amdgcn_logf(s1 * 0.25f));
            __builtin_nontemporal_store(o, (v2f*)(ob + r * WO + lane * 2));
        }
    }
}

extern "C" void kernel_launch(void* const* d_in, const int* in_sizes, int n_in,
                              void* d_out, int out_size, void* d_ws, size_t ws_size,
                              hipStream_t stream) {
    const float* x  = (const float*)d_in[0];  // (B,C,H,W) fp32
    const float* pp = (const float*)d_in[1];  // (C,) fp32
    const float* cc = (const float*)d_in[2];  // (C,2,2) fp32
    float* out = (float*)d_out;               // (B,C,Ho,Wo) fp32

    const int BC = in_sizes[0] / (H_ * W_);   // B*C = 4096
    lp_pool2d_async_kernel<<<dim3(BC), dim3(BDIM), 0, stream>>>(x, pp, cc, out);
}